// GCNLayer_63393717289266
// MI455X (gfx1250) — compile-verified
//
#include <hip/hip_runtime.h>

typedef __attribute__((ext_vector_type(2))) float v2f;
typedef __attribute__((ext_vector_type(8))) float v8f;

#define IN_F  256
#define OUT_F 128
#define KSTEPS (IN_F / 4)          // 64 WMMA K-steps

// ---------------------------------------------------------------------------
// zero the edge accumulator
// ---------------------------------------------------------------------------
__global__ void zero_kernel(float* __restrict__ p, int n) {
    int i      = blockIdx.x * blockDim.x + threadIdx.x;
    int stride = gridDim.x * blockDim.x;
    for (; i < n; i += stride) p[i] = 0.0f;
}

// ---------------------------------------------------------------------------
// C = A[.,256] @ W[256, cols] + b (+ bias2), fp32 WMMA 16x16x4.
// Block = 256 threads = 8 waves. The block stages one 256x128 W tile into
// LDS (128 KB of the 320 KB WGP LDS), pre-packed so each lane's B-fragment
// (K=kk, kk+1) is one 8-byte LDS element -> single conflict-free ds_load_b64
// per fragment. Each wave computes a 16x128 output tile (8 v8f accumulators),
// A fragment is software-pipelined one K-step ahead.
// Fragment layouts per ISA: A 16x4 f32 -> lanes 0-15 hold K=k,k+1, lanes
// 16-31 hold K=k+2,k+3 (B symmetric). C/D: row = r + 8*(lane>=16), col = lane&15.
// ---------------------------------------------------------------------------
__global__ void __launch_bounds__(256)
gemm_wmma_f32(const float* __restrict__ A,
              const float* __restrict__ W,
              const float* __restrict__ b,
              const float* __restrict__ bias2,
              float* __restrict__ C,
              int nrows, int cols) {
    __shared__ float sWp[IN_F * 128];              // 128 KB, packed fragments

    const int tid  = threadIdx.x;
    const int lane = tid & 31;
    const int wv   = tid >> 5;                     // wave 0..7
    const int lo   = lane & 15;
    const int hi   = lane >> 4;                    // 0 or 1
    const int n0   = blockIdx.y * 128;

    // ---- stage + pack W tile: coalesced global reads, permuted LDS writes
    for (int idx = tid; idx < IN_F * 128; idx += 256) {
        const int k = idx >> 7;                    // 0..255 (K row)
        const int c = idx & 127;                   // 0..127 (col in tile)
        const float w = W[(size_t)k * cols + (n0 + c)];
        const int kstep = k >> 2;
        const int krem  = k & 3;
        const int lane2 = (krem >> 1) * 16 + (c & 15);
        const int j     = c >> 4;
        sWp[((((kstep << 3) + j) << 5) + lane2) * 2 + (krem & 1)] = w;
    }
    __syncthreads();

    // ---- compute: one 16x128 tile per wave
    const int m0 = blockIdx.x * 128 + wv * 16;
    v8f acc[8] = {};

    int arow = m0 + lo;
    if (arow >= nrows) arow = nrows - 1;           // clamp loads only
    const float* Arow = A + (size_t)arow * IN_F + hi * 2;

    v2f a = *(const v2f*)(Arow);                   // K-step 0 fragment
    for (int ks = 0; ks < KSTEPS; ++ks) {
        const v2f a_cur = a;
        if (ks + 1 < KSTEPS)                       // pipeline next A fragment
            a = *(const v2f*)(Arow + (ks + 1) * 4);
        const v2f* bp = (const v2f*)&sWp[((ks << 3) << 5) * 2] + lane;
#pragma unroll
        for (int j = 0; j < 8; ++j) {
            const v2f bf = bp[j << 5];             // ds_load_b64, no conflicts
            acc[j] = __builtin_amdgcn_wmma_f32_16x16x4_f32(
                false, a_cur, false, bf, (short)0, acc[j], false, false);
        }
    }

    // ---- epilogue: bias add + store
#pragma unroll
    for (int j = 0; j < 8; ++j) {
        const int col  = n0 + j * 16 + lo;
        const float ba = b[col] + (bias2 ? bias2[col] : 0.0f);
#pragma unroll
        for (int r = 0; r < 8; ++r) {
            const int row = m0 + r + 8 * hi;
            if (row < nrows) C[(size_t)row * cols + col] = acc[j][r] + ba;
        }
    }
}

// ---------------------------------------------------------------------------
// One wave per edge (grid-stride). Lane handles 4 consecutive output floats.
// msg = degrees @ W_edge + sum_i mask_i * Hmsg[src, i], scaled by norm[src],
// atomically accumulated into acc[dst]. degrees are loaded once (lane<6) and
// broadcast with wave32 shuffles.
// ---------------------------------------------------------------------------
__global__ void edge_kernel(const float* __restrict__ degrees,
                            const int*   __restrict__ src,
                            const int*   __restrict__ dst,
                            const float* __restrict__ norm,
                            const float* __restrict__ Wedge,
                            const float* __restrict__ Hmsg,
                            float*       __restrict__ acc,
                            int E) {
    __shared__ float sWe[6 * OUT_F];               // 3 KB
    for (int i = threadIdx.x; i < 6 * OUT_F; i += blockDim.x) sWe[i] = Wedge[i];
    __syncthreads();

    const int lane   = threadIdx.x & 31;
    const int wave   = (blockIdx.x * blockDim.x + threadIdx.x) >> 5;
    const int nwaves = (gridDim.x * blockDim.x) >> 5;
    const int f0     = lane * 4;                   // 32 lanes * 4 = 128 cols

    for (int e = wave; e < E; e += nwaves) {
        const int   s   = src[e];
        const int   d   = dst[e];
        const float nrm = norm[s];
        const float* hb = Hmsg + (size_t)s * (6 * OUT_F);

        // one lane-strided load of the 6 degrees, then wave32 broadcast
        float dlane = 0.0f;
        if (lane < 6) dlane = degrees[(size_t)e * 6 + lane];

        float m0 = 0.f, m1 = 0.f, m2 = 0.f, m3 = 0.f;
#pragma unroll
        for (int i = 0; i < 6; ++i) {
            const float di = __shfl(dlane, i, 32);
            m0 += di * sWe[i * OUT_F + f0 + 0];
            m1 += di * sWe[i * OUT_F + f0 + 1];
            m2 += di * sWe[i * OUT_F + f0 + 2];
            m3 += di * sWe[i * OUT_F + f0 + 3];
            if (di > 0.0f) {                       // mask = degrees > 0
                const float4 hv = *(const float4*)(hb + i * OUT_F + f0);
                m0 += hv.x; m1 += hv.y; m2 += hv.z; m3 += hv.w;
            }
        }
        float* ap = acc + (size_t)d * OUT_F + f0;
        atomicAdd(ap + 0, m0 * nrm);
        atomicAdd(ap + 1, m1 * nrm);
        atomicAdd(ap + 2, m2 * nrm);
        atomicAdd(ap + 3, m3 * nrm);
    }
}

// ---------------------------------------------------------------------------
// out = relu(acc * norm[row] + out)   (out already holds h@W_self+b_self+bias)
// ---------------------------------------------------------------------------
__global__ void finish_kernel(const float* __restrict__ acc,
                              const float* __restrict__ norm,
                              float* __restrict__ out, int total) {
    int i      = blockIdx.x * blockDim.x + threadIdx.x;
    int stride = gridDim.x * blockDim.x;
    for (; i < total; i += stride) {
        const float v = acc[i] * norm[i >> 7] + out[i];
        out[i] = v > 0.0f ? v : 0.0f;
    }
}

// ---------------------------------------------------------------------------
extern "C" void kernel_launch(void* const* d_in, const int* in_sizes, int n_in,
                              void* d_out, int out_size, void* d_ws, size_t ws_size,
                              hipStream_t stream) {
    const float* h       = (const float*)d_in[0];
    const float* degrees = (const float*)d_in[1];
    const float* norm    = (const float*)d_in[2];
    const int*   src     = (const int*)  d_in[3];
    const int*   dst     = (const int*)  d_in[4];
    const float* W_self  = (const float*)d_in[5];
    const float* b_self  = (const float*)d_in[6];
    const float* W_node  = (const float*)d_in[7];
    const float* b_node  = (const float*)d_in[8];
    const float* W_edge  = (const float*)d_in[9];
    const float* bias    = (const float*)d_in[10];
    float* out = (float*)d_out;

    const int N = in_sizes[2];          // norm: [N,1]
    const int E = in_sizes[3];          // src:  [E]

    float* Hmsg = (float*)d_ws;                         // N * 768 floats
    float* acc  = Hmsg + (size_t)N * 6 * OUT_F;         // N * 128 floats

    // 1) zero edge accumulator
    zero_kernel<<<1024, 256, 0, stream>>>(acc, N * OUT_F);

    // 2) Hmsg = h @ W_node + b_node           (19.7 GFLOP, fp32 WMMA)
    dim3 g1((N + 127) / 128, (6 * OUT_F) / 128);
    gemm_wmma_f32<<<g1, 256, 0, stream>>>(h, W_node, b_node, nullptr, Hmsg, N, 6 * OUT_F);

    // 3) out = h @ W_self + b_self + bias     (3.3 GFLOP, fp32 WMMA)
    dim3 g2((N + 127) / 128, 1);
    gemm_wmma_f32<<<g2, 256, 0, stream>>>(h, W_self, b_self, bias, out, N, OUT_F);

    // 4) edge gather + scatter-add (L2-resident Hmsg: 153.6 MB < 192 MB L2)
    edge_kernel<<<2048, 256, 0, stream>>>(degrees, src, dst, norm, W_edge, Hmsg, acc, E);

    // 5) fused dst-norm + residual + relu
    finish_kernel<<<4096, 256, 0, stream>>>(acc, norm, out, N * OUT_F);
}